// SimpleNodeBlock_84198538871458
// MI455X (gfx1250) — compile-verified
//
#include <hip/hip_runtime.h>
#include <stdint.h>
#include <math.h>

typedef float v2f __attribute__((ext_vector_type(2)));
typedef float v4f __attribute__((ext_vector_type(4)));
typedef float v8f __attribute__((ext_vector_type(8)));

#define N_NODES 50000
#define NTILES  3125                 // 50000 / 16 exactly
#define KSTEPS  652                  // 2608 / 4
#define WCATP_ELEMS (KSTEPS * 256)   // 166912 floats: [kstep][g(2)][lane(32)][4]
#define W3P_ELEMS   (32 * 16 * 128)  // 65536 floats:  [u*16+ks][lane(32)][4]

// ---------------------------------------------------------------------------
// Pack weights into WMMA B-fragment-major layout (constants folded in).
// B-frag convention for V_WMMA_F32_16X16X4_F32 (4x16 B):
//   lane L, element j holds  B[K = 2*(L>>4) + j][col = L&15]
// Each lane's 4 consecutive floats cover two adjacent 16-col tiles
// (e = {tileLo j0, tileLo j1, tileHi j0, tileHi j1}) -> one b128 load / 2 frags.
// ---------------------------------------------------------------------------
__global__ void pack_weights_kernel(const float* __restrict__ W1,
                                    const float* __restrict__ W2,
                                    const float* __restrict__ W3,
                                    const float* __restrict__ W4,
                                    const float* __restrict__ W5,
                                    float* __restrict__ WcatP,
                                    float* __restrict__ W3P) {
    const float c0  = (float)(1.0 / sqrt(2608.0));               // C0
    const float c0i = (float)(1.0 / (sqrt(2608.0) * sqrt(3.0))); // C0/sqrt(3)
    const float c1i = (float)(1.0 / sqrt(2048.0));               // C1/sqrt(3)
    int idx = blockIdx.x * blockDim.x + threadIdx.x;
    if (idx < WCATP_ELEMS) {
        int kp  = idx >> 8;              // /256
        int rem = idx & 255;
        int g   = rem >> 7;              // tile pair (cols 0-31 or 32-63)
        int l4  = rem & 127;
        int L   = l4 >> 2;               // lane 0..31
        int e   = l4 & 3;
        int nt  = g * 2 + (e >> 1);
        int j   = e & 1;
        int p   = 4 * kp + 2 * (L >> 4) + j;   // concatenated feature row
        int c   = nt * 16 + (L & 15);          // output channel
        float v;
        if      (p < 2016) v = c0  * W1[p * 64 + c];
        else if (p < 2080) v = c0  * W2[(p - 2016) * 64 + c];
        else if (p < 2576) v = c0i * W4[(p - 2080) * 64 + c];
        else               v = c0i * W5[(p - 2576) * 64 + c];
        WcatP[idx] = v;
    } else if (idx < WCATP_ELEMS + W3P_ELEMS) {
        int t      = idx - WCATP_ELEMS;
        int blk    = t >> 7;             // u*16 + ks
        int within = t & 127;
        int L  = within >> 2;            // lane 0..31
        int e  = within & 3;
        int ks = blk & 15;
        int u  = blk >> 4;
        int wt = e >> 1;
        int j  = e & 1;
        int v  = ks * 4 + 2 * (L >> 4) + j;    // K index over x0 channels
        int w  = wt * 16 + (L & 15);           // output w channel
        W3P[t] = c1i * W3[u * 2048 + v * 32 + w];
    }
}

static __device__ __forceinline__ v8f wmma_f32(v2f a, v2f b, v8f c) {
    // 8-arg form: (neg_a, A, neg_b, B, c_mod, C, reuse_a, reuse_b)
    return __builtin_amdgcn_wmma_f32_16x16x4_f32(false, a, false, b, (short)0, c,
                                                 false, false);
}
static __device__ __forceinline__ v2f lo2(v4f q) {
    return __builtin_shufflevector(q, q, 0, 1);
}
static __device__ __forceinline__ v2f hi2(v4f q) {
    return __builtin_shufflevector(q, q, 2, 3);
}

// ---------------------------------------------------------------------------
// Main kernel: 1 wave handles 16 nodes. blockDim = 128 (4 waves, wave32).
// ---------------------------------------------------------------------------
__global__ __launch_bounds__(128) void tensor_square_kernel(
    const float* __restrict__ nf, const float* __restrict__ nm,
    const float* __restrict__ WcatP, const float* __restrict__ W3P,
    float* __restrict__ out)
{
    __shared__ __align__(4) uint16_t idx0[2016]; // (u | v<<8), triu(64, k=1)
    __shared__ __align__(4) uint16_t idx1[496];  // (u | v<<8), triu(32, k=1)
    __shared__ float xs[4][2560];   // per-wave: x0[16][64] + x1SoA[3][16][32]

    const int tid = threadIdx.x;
    // Build pair-index tables once per workgroup.
    if (tid < 64) {
        int u = tid;
        int base = u * 63 - (u * (u - 1)) / 2;
        for (int v = u + 1; v < 64; ++v)
            idx0[base + (v - u - 1)] = (uint16_t)(u | (v << 8));
    } else if (tid < 96) {
        int u = tid - 64;
        int base = u * 31 - (u * (u - 1)) / 2;
        for (int v = u + 1; v < 32; ++v)
            idx1[base + (v - u - 1)] = (uint16_t)(u | (v << 8));
    }
    __syncthreads();

    const int wave = tid >> 5;
    const int lane = tid & 31;
    const int tile = blockIdx.x * 4 + wave;
    if (tile >= NTILES) return;         // wave-uniform guard (EXEC stays all-1)

    float* xw = xs[wave];

    // Stage x = nf + nm for this wave's 16 nodes into LDS (x1 stored SoA).
    {
        const int gbase = tile * 16 * 160;
        for (int q = lane; q < 2560; q += 32) {
            int n = q / 160;
            int f = q - n * 160;
            float v = nf[gbase + q] + nm[gbase + q];
            int dst;
            if (f < 64) dst = n * 64 + f;
            else { int g = f - 64; int u = g / 3; int i = g - 3 * u;
                   dst = 1024 + i * 512 + n * 32 + u; }
            xw[dst] = v;
        }
    }
    asm volatile("s_wait_dscnt 0" ::: "memory"); // per-wave LDS staging visible

    const int node = lane & 15;   // A-matrix row (this lane's node) / D col
    const int half = lane >> 4;   // which K/M half this lane covers
    const float* x0p  = xw + node * 64;
    const float* x1p  = xw + 1024 + node * 32;   // plane stride 512
    const float* wrow = WcatP + lane * 4;
    const uint32_t* idx0d = (const uint32_t*)idx0;  // dword = 2 packed entries
    const uint32_t* idx1d = (const uint32_t*)idx1;

    v8f a0 = {0,0,0,0,0,0,0,0}, a1 = a0, a2 = a0, a3 = a0;

    // Software-pipelined B fragments: cq0 covers col-tiles 0,1; cq1 tiles 2,3.
    v4f cq0 = *(const v4f*)(wrow);
    v4f cq1 = *(const v4f*)(wrow + 128);

    // ---- Phase A: out0 = F(2608) @ Wcat(2608x64), A-frags built on the fly ----
    // Segment 1: pair0 (steps 0..503). A side pipelined depth-2 on the index
    // dword, depth-1 on operands, so no LDS latency sits on the WMMA path.
    {
        unsigned edc = idx0d[half];          // indices for kp=0
        unsigned edn = idx0d[2 + half];      // indices for kp=1
        float cxu = x0p[edc & 255],         cxv = x0p[(edc >> 8) & 255];
        float cyu = x0p[(edc >> 16) & 255], cyv = x0p[edc >> 24];
        #pragma unroll 1
        for (int kp = 0; kp < 504; ++kp) {
            const float* wn = wrow + (kp + 1) * 256;       // B frags for kp+1
            v4f nq0 = *(const v4f*)(wn);
            v4f nq1 = *(const v4f*)(wn + 128);
            unsigned ed2 = idx0d[(kp + 2) * 2 + half];     // idx for kp+2 (tail
                                                           // reads are benign)
            float nxu = x0p[edn & 255],         nxv = x0p[(edn >> 8) & 255];
            float nyu = x0p[(edn >> 16) & 255], nyv = x0p[edn >> 24];
            v2f a; a.x = cxu * cxv; a.y = cyu * cyv;
            a0 = wmma_f32(a, lo2(cq0), a0);
            a1 = wmma_f32(a, hi2(cq0), a1);
            a2 = wmma_f32(a, lo2(cq1), a2);
            a3 = wmma_f32(a, hi2(cq1), a3);
            cq0 = nq0; cq1 = nq1;
            cxu = nxu; cxv = nxv; cyu = nyu; cyv = nyv;
            edn = ed2;
        }
    }
    // Segment 2: sq0  (steps 504..519) -- short, unpipelined A side.
    #pragma unroll 1
    for (int kp = 504; kp < 520; ++kp) {
        const float* wn = wrow + (kp + 1) * 256;
        v4f nq0 = *(const v4f*)(wn);
        v4f nq1 = *(const v4f*)(wn + 128);
        int jj = kp * 4 + half * 2 - 2016;
        v2f xv = *(const v2f*)(x0p + jj);
        v2f a; a.x = xv.x * xv.x; a.y = xv.y * xv.y;
        a0 = wmma_f32(a, lo2(cq0), a0);
        a1 = wmma_f32(a, hi2(cq0), a1);
        a2 = wmma_f32(a, lo2(cq1), a2);
        a3 = wmma_f32(a, hi2(cq1), a3);
        cq0 = nq0; cq1 = nq1;
    }
    // Segment 3: pair1 = dot3(x1u, x1v) (steps 520..643), A side pipelined.
    {
        unsigned edc = idx1d[half];
        unsigned edn = idx1d[2 + half];
        int u0 = edc & 255, v0 = (edc >> 8) & 255;
        int u1 = (edc >> 16) & 255, v1 = edc >> 24;
        float c0u = x1p[u0],        c0v = x1p[v0];
        float c1u = x1p[512 + u0],  c1v = x1p[512 + v0];
        float c2u = x1p[1024 + u0], c2v = x1p[1024 + v0];
        float d0u = x1p[u1],        d0v = x1p[v1];
        float d1u = x1p[512 + u1],  d1v = x1p[512 + v1];
        float d2u = x1p[1024 + u1], d2v = x1p[1024 + v1];
        #pragma unroll 1
        for (int kp = 520; kp < 644; ++kp) {
            const float* wn = wrow + (kp + 1) * 256;
            v4f nq0 = *(const v4f*)(wn);
            v4f nq1 = *(const v4f*)(wn + 128);
            unsigned ed2 = idx1d[(kp - 520 + 2) * 2 + half]; // tail reads benign
            int nu0 = edn & 255, nv0 = (edn >> 8) & 255;
            int nu1 = (edn >> 16) & 255, nv1 = edn >> 24;
            float n0u = x1p[nu0],        n0v = x1p[nv0];
            float n1u = x1p[512 + nu0],  n1v = x1p[512 + nv0];
            float n2u = x1p[1024 + nu0], n2v = x1p[1024 + nv0];
            float m0u = x1p[nu1],        m0v = x1p[nv1];
            float m1u = x1p[512 + nu1],  m1v = x1p[512 + nv1];
            float m2u = x1p[1024 + nu1], m2v = x1p[1024 + nv1];
            v2f a;
            a.x = c0u * c0v + c1u * c1v + c2u * c2v;
            a.y = d0u * d0v + d1u * d1v + d2u * d2v;
            a0 = wmma_f32(a, lo2(cq0), a0);
            a1 = wmma_f32(a, hi2(cq0), a1);
            a2 = wmma_f32(a, lo2(cq1), a2);
            a3 = wmma_f32(a, hi2(cq1), a3);
            cq0 = nq0; cq1 = nq1;
            c0u = n0u; c0v = n0v; c1u = n1u; c1v = n1v; c2u = n2u; c2v = n2v;
            d0u = m0u; d0v = m0v; d1u = m1u; d1v = m1v; d2u = m2u; d2v = m2v;
            edn = ed2;
        }
    }
    // Segment 4: diag1 = dot3(x1j, x1j)  (steps 644..651)
    // (final B prefetch lands in the adjacent W3P region of d_ws -- valid)
    #pragma unroll 1
    for (int kp = 644; kp < 652; ++kp) {
        const float* wn = wrow + (kp + 1) * 256;
        v4f nq0 = *(const v4f*)(wn);
        v4f nq1 = *(const v4f*)(wn + 128);
        int j0 = kp * 4 + half * 2 - 2576;
        float p0a = x1p[j0],     p0b = x1p[512 + j0],     p0c = x1p[1024 + j0];
        float p1a = x1p[j0 + 1], p1b = x1p[512 + j0 + 1], p1c = x1p[1024 + j0 + 1];
        v2f a;
        a.x = p0a * p0a + p0b * p0b + p0c * p0c;
        a.y = p1a * p1a + p1b * p1b + p1c * p1c;
        a0 = wmma_f32(a, lo2(cq0), a0);
        a1 = wmma_f32(a, hi2(cq0), a1);
        a2 = wmma_f32(a, lo2(cq1), a2);
        a3 = wmma_f32(a, hi2(cq1), a3);
        cq0 = nq0; cq1 = nq1;
    }
    // Store out0: lane holds rows (r + 8*half), col = node (+ nt*16).
    {
        float* orow = out + (tile * 16 + half * 8) * 160 + node;
        #pragma unroll
        for (int r = 0; r < 8; ++r) {
            orow[r * 160     ] = a0[r];
            orow[r * 160 + 16] = a1[r];
            orow[r * 160 + 32] = a2[r];
            orow[r * 160 + 48] = a3[r];
        }
    }

    // ---- Phase B: B[n,u,w] = x0 @ W3r, then out1[n,w,i] += x1[n,u,i]*B ----
    v2f xa[16];                       // whole x0 tile as A-fragments (K=64)
    #pragma unroll
    for (int ks = 0; ks < 16; ++ks)
        xa[ks] = *(const v2f*)(x0p + ks * 4 + half * 2);

    float o1[2][3][8];
    #pragma unroll
    for (int wt = 0; wt < 2; ++wt)
        for (int i = 0; i < 3; ++i)
            for (int r = 0; r < 8; ++r) o1[wt][i][r] = 0.0f;

    const float* w3row = W3P + lane * 4;
    const float* x1q   = xw + 1024 + half * 256;  // + i*512 + r*32 + u

    #pragma unroll 1
    for (int u = 0; u < 32; ++u) {
        v8f b0 = {0,0,0,0,0,0,0,0}, b1 = b0;
        const float* wb = w3row + u * 2048;       // 16 ks * 128
        #pragma unroll
        for (int ks = 0; ks < 16; ++ks) {
            v4f q = *(const v4f*)(wb + ks * 128); // both wt tiles in one b128
            b0 = wmma_f32(xa[ks], lo2(q), b0);
            b1 = wmma_f32(xa[ks], hi2(q), b1);
        }
        #pragma unroll
        for (int i = 0; i < 3; ++i) {
            const float* xv = x1q + i * 512 + u;
            #pragma unroll
            for (int r = 0; r < 8; ++r) {
                float x1v = xv[r * 32];
                o1[0][i][r] = fmaf(x1v, b0[r], o1[0][i][r]);
                o1[1][i][r] = fmaf(x1v, b1[r], o1[1][i][r]);
            }
        }
    }
    // Store out1: flat col = 64 + w*3 + i, w = wt*16 + node.
    {
        float* orow = out + (tile * 16 + half * 8) * 160 + 64 + node * 3;
        #pragma unroll
        for (int wt = 0; wt < 2; ++wt)
            for (int i = 0; i < 3; ++i)
                for (int r = 0; r < 8; ++r)
                    orow[r * 160 + wt * 48 + i] = o1[wt][i][r];
    }
}

extern "C" void kernel_launch(void* const* d_in, const int* in_sizes, int n_in,
                              void* d_out, int out_size, void* d_ws, size_t ws_size,
                              hipStream_t stream) {
    (void)in_sizes; (void)n_in; (void)out_size; (void)ws_size;
    const float* nf = (const float*)d_in[0];
    const float* nm = (const float*)d_in[1];
    const float* W1 = (const float*)d_in[2];
    const float* W2 = (const float*)d_in[3];
    const float* W3 = (const float*)d_in[4];
    const float* W4 = (const float*)d_in[5];
    const float* W5 = (const float*)d_in[6];

    float* WcatP = (float*)d_ws;                 // 166912 floats
    float* W3P   = WcatP + WCATP_ELEMS;          // 65536 floats (~0.93 MB total)

    const int packTotal = WCATP_ELEMS + W3P_ELEMS;
    pack_weights_kernel<<<(packTotal + 255) / 256, 256, 0, stream>>>(
        W1, W2, W3, W4, W5, WcatP, W3P);

    tensor_square_kernel<<<(NTILES + 3) / 4, 128, 0, stream>>>(
        nf, nm, WcatP, W3P, (float*)d_out);
}